// PointNetFeaturePropagation_77257871720537
// MI455X (gfx1250) — compile-verified
//
#include <hip/hip_runtime.h>

typedef __attribute__((ext_vector_type(2))) float v2f;
typedef __attribute__((ext_vector_type(8))) float v8f;

#define B_   8
#define N_   8192
#define S_   2048
#define C1_  128
#define C2_  256
#define INCH 384          // C1 + C2
#define H1   256
#define H2   128
#define M_   (B_ * N_)    // 65536 rows
#define EPS_INTERP 1e-8f
#define EPS_BN     1e-5f

// ---------------------------------------------------------------------------
// Kernel 1: 3-NN + inverse-distance interpolation + concat.
// One thread per query point. xyz2[b] (24KB) staged in LDS.
// Writes X column-major: Xc[k][row], k in [0,384), row = b*N + n.
// ---------------------------------------------------------------------------
__global__ __launch_bounds__(256) void knn_interp_concat(
    const float* __restrict__ xyz1, const float* __restrict__ xyz2,
    const float* __restrict__ points1, const float* __restrict__ points2,
    float* __restrict__ Xc)
{
    __shared__ float sx[S_], sy[S_], sz[S_];
    const int tid = threadIdx.x;
    const int blocksPerBatch = N_ / 256;
    const int b = blockIdx.x / blocksPerBatch;
    const int n = (blockIdx.x % blocksPerBatch) * 256 + tid;

    const float* x2 = xyz2 + (size_t)b * 3 * S_;
    for (int s = tid; s < S_; s += 256) {
        sx[s] = x2[s];
        sy[s] = x2[S_ + s];
        sz[s] = x2[2 * S_ + s];
    }
    __syncthreads();

    const float px = xyz1[(size_t)b * 3 * N_ + n];
    const float py = xyz1[(size_t)b * 3 * N_ + N_ + n];
    const float pz = xyz1[(size_t)b * 3 * N_ + 2 * N_ + n];

    float d0 = 3.4e38f, d1 = 3.4e38f, d2 = 3.4e38f;
    int   i0 = 0, i1 = 0, i2 = 0;
    for (int s = 0; s < S_; ++s) {
        const float dx = px - sx[s], dy = py - sy[s], dz = pz - sz[s];
        const float d = dx * dx + dy * dy + dz * dz;
        if (d < d0)      { d2 = d1; i2 = i1; d1 = d0; i1 = i0; d0 = d; i0 = s; }
        else if (d < d1) { d2 = d1; i2 = i1; d1 = d;  i1 = s; }
        else if (d < d2) { d2 = d;  i2 = s; }
    }
    const float r0 = 1.f / (d0 + EPS_INTERP);
    const float r1 = 1.f / (d1 + EPS_INTERP);
    const float r2 = 1.f / (d2 + EPS_INTERP);
    const float inv = 1.f / (r0 + r1 + r2);
    const float w0 = r0 * inv, w1 = r1 * inv, w2 = r2 * inv;

    const size_t row = (size_t)b * N_ + n;

    // rows 0..127 of Xc: points1 (coalesced read & write: thread-per-n)
    const float* p1 = points1 + (size_t)b * C1_ * N_ + n;
    for (int c = 0; c < C1_; ++c)
        Xc[(size_t)c * M_ + row] = p1[(size_t)c * N_];

    // rows 128..383: interpolated points2 features
    const float* p2 = points2 + (size_t)b * C2_ * S_;
    for (int c = 0; c < C2_; ++c) {
        const float* pc = p2 + (size_t)c * S_;
        Xc[(size_t)(C1_ + c) * M_ + row] = w0 * pc[i0] + w1 * pc[i1] + w2 * pc[i2];
    }
}

// ---------------------------------------------------------------------------
// Kernel 2: layer-1 GEMM: Y1[row][o] = sum_k Xc[k][row]*W1[o][k] + b1[o]
// fp32 WMMA 16x16x4. Wave tile = 16M x 128N (8 accumulators).
// Block = 8 waves -> 128M x 128N. grid = (M/128, H1/128).
// A frag: lanes 0-15 hold K=k,k+1 ; lanes 16-31 hold K=k+2,k+3 (M = lane&15).
// B frag: same K split, N = lane&15.  C/D: N = lane&15, M = vgpr + 8*(lane>>4).
// ---------------------------------------------------------------------------
__global__ __launch_bounds__(256) void gemm1_wmma(
    const float* __restrict__ Xc, const float* __restrict__ W1,
    const float* __restrict__ b1, float* __restrict__ Y1)
{
    const int lane = threadIdx.x & 31;
    const int wave = threadIdx.x >> 5;
    const int rowA  = blockIdx.x * 128 + wave * 16 + (lane & 15);
    const int nBase = blockIdx.y * 128;
    const int kSel  = (lane >> 4) * 2;
    const int nCol  = nBase + (lane & 15);

    v8f acc[8] = {};

    for (int k = 0; k < INCH; k += 4) {
        const int kb = k + kSel;
        v2f A;
        A.x = Xc[(size_t)kb * M_ + rowA];
        A.y = Xc[(size_t)(kb + 1) * M_ + rowA];
        __builtin_prefetch(Xc + (size_t)(kb + 16) * M_ + rowA, 0, 1);
#pragma unroll
        for (int j = 0; j < 8; ++j) {
            const float* wp = W1 + (size_t)(nCol + j * 16) * INCH + kb;
            v2f Bm;
            Bm.x = wp[0];
            Bm.y = wp[1];
            acc[j] = __builtin_amdgcn_wmma_f32_16x16x4_f32(
                false, A, false, Bm, (short)0, acc[j], false, false);
        }
    }

    const int mBase = blockIdx.x * 128 + wave * 16 + (lane >> 4) * 8;
#pragma unroll
    for (int j = 0; j < 8; ++j) {
        const int col = nCol + j * 16;
        const float bias = b1[col];
#pragma unroll
        for (int g = 0; g < 8; ++g)
            Y1[(size_t)(mBase + g) * H1 + col] = acc[j][g] + bias;
    }
}

// ---------------------------------------------------------------------------
// Kernel 3: per-channel sum / sum-of-squares (BN training stats).
// blockDim.x == cols; each block reduces 256 rows; coalesced loads.
// ---------------------------------------------------------------------------
__global__ void colstats(const float* __restrict__ Y, int cols,
                         float* __restrict__ sum, float* __restrict__ sumsq)
{
    const int c = threadIdx.x;
    const size_t r0 = (size_t)blockIdx.x * 256;
    float s = 0.f, s2 = 0.f;
    for (int r = 0; r < 256; ++r) {
        const float v = Y[(r0 + r) * cols + c];
        s += v;
        s2 += v * v;
    }
    atomicAdd(&sum[c], s);
    atomicAdd(&sumsq[c], s2);
}

// ---------------------------------------------------------------------------
// Kernel 4: fold BN stats + gamma/beta into per-channel scale/shift.
// ---------------------------------------------------------------------------
__global__ void bnparams(const float* __restrict__ sum, const float* __restrict__ sumsq,
                         const float* __restrict__ g, const float* __restrict__ be,
                         float* __restrict__ sc, float* __restrict__ sh, int cols)
{
    const int c = threadIdx.x;
    if (c >= cols) return;
    const float invCnt = 1.f / (float)M_;
    const float mean = sum[c] * invCnt;
    const float var  = sumsq[c] * invCnt - mean * mean;
    const float s    = g[c] * rsqrtf(var + EPS_BN);
    sc[c] = s;
    sh[c] = be[c] - mean * s;
}

// ---------------------------------------------------------------------------
// Kernel 5: layer-2 GEMM with fused BN1-affine + ReLU on the A loads.
// A row = relu(Y1*sc1+sh1); Y2[row][o] = A·W2[o][:] + b2[o].
// Wave tile 16M x 128N covers all H2 columns; grid = (M/128, 1).
// ---------------------------------------------------------------------------
__global__ __launch_bounds__(256) void gemm2_wmma(
    const float* __restrict__ Y1, const float* __restrict__ W2,
    const float* __restrict__ b2, const float* __restrict__ sc1,
    const float* __restrict__ sh1, float* __restrict__ Y2)
{
    const int lane = threadIdx.x & 31;
    const int wave = threadIdx.x >> 5;
    const int rowA = blockIdx.x * 128 + wave * 16 + (lane & 15);
    const int kSel = (lane >> 4) * 2;
    const int nCol = lane & 15;

    v8f acc[8] = {};

    for (int k = 0; k < H1; k += 4) {
        const int kb = k + kSel;
        const float* yp = Y1 + (size_t)rowA * H1 + kb;   // contiguous float2 per lane
        v2f A;
        A.x = fmaxf(yp[0] * sc1[kb]     + sh1[kb],     0.f);
        A.y = fmaxf(yp[1] * sc1[kb + 1] + sh1[kb + 1], 0.f);
#pragma unroll
        for (int j = 0; j < 8; ++j) {
            const float* wp = W2 + (size_t)(nCol + j * 16) * H1 + kb;
            v2f Bm;
            Bm.x = wp[0];
            Bm.y = wp[1];
            acc[j] = __builtin_amdgcn_wmma_f32_16x16x4_f32(
                false, A, false, Bm, (short)0, acc[j], false, false);
        }
    }

    const int mBase = blockIdx.x * 128 + wave * 16 + (lane >> 4) * 8;
#pragma unroll
    for (int j = 0; j < 8; ++j) {
        const int col = nCol + j * 16;
        const float bias = b2[col];
#pragma unroll
        for (int g = 0; g < 8; ++g)
            Y2[(size_t)(mBase + g) * H2 + col] = acc[j][g] + bias;
    }
}

// ---------------------------------------------------------------------------
// Kernel 6: BN2-affine + ReLU + transpose [65536,128] -> [B,128,N].
// 32x32 LDS tile, coalesced loads and stores.
// ---------------------------------------------------------------------------
__global__ __launch_bounds__(256) void bn_relu_transpose_out(
    const float* __restrict__ Y2, const float* __restrict__ sc2,
    const float* __restrict__ sh2, float* __restrict__ out)
{
    __shared__ float tile[32][33];
    const int tx = threadIdx.x & 31;
    const int ty = threadIdx.x >> 5;   // 0..7
    const int rowBase = blockIdx.x * 32;
    const int colBase = blockIdx.y * 32;

#pragma unroll
    for (int i = 0; i < 4; ++i) {
        const int r = rowBase + ty + i * 8;
        const int c = colBase + tx;
        const float v = Y2[(size_t)r * H2 + c];
        tile[ty + i * 8][tx] = fmaxf(v * sc2[c] + sh2[c], 0.f);
    }
    __syncthreads();

    const int b  = rowBase / N_;
    const int nb = rowBase % N_;       // all 32 rows within one batch (8192 % 32 == 0)
#pragma unroll
    for (int i = 0; i < 4; ++i) {
        const int c = colBase + ty + i * 8;
        out[(size_t)b * H2 * N_ + (size_t)c * N_ + nb + tx] = tile[tx][ty + i * 8];
    }
}

// ---------------------------------------------------------------------------
extern "C" void kernel_launch(void* const* d_in, const int* in_sizes, int n_in,
                              void* d_out, int out_size, void* d_ws, size_t ws_size,
                              hipStream_t stream)
{
    const float* xyz1    = (const float*)d_in[0];
    const float* xyz2    = (const float*)d_in[1];
    const float* points1 = (const float*)d_in[2];
    const float* points2 = (const float*)d_in[3];
    const float* W1  = (const float*)d_in[4];
    const float* b1  = (const float*)d_in[5];
    const float* g1  = (const float*)d_in[6];
    const float* be1 = (const float*)d_in[7];
    const float* W2  = (const float*)d_in[8];
    const float* b2  = (const float*)d_in[9];
    const float* g2  = (const float*)d_in[10];
    const float* be2 = (const float*)d_in[11];
    float* out = (float*)d_out;

    // Workspace layout (floats): Xc | Y1 | Y2 | stats | params   (~201 MB)
    float* ws = (float*)d_ws;
    float* Xc = ws;                                 // 384 * 65536
    float* Y1 = Xc + (size_t)INCH * M_;             // 65536 * 256
    float* Y2 = Y1 + (size_t)M_ * H1;               // 65536 * 128
    float* stats = Y2 + (size_t)M_ * H2;            // sum1,sq1,sum2,sq2 contiguous
    float* sum1 = stats;
    float* sq1  = sum1 + H1;
    float* sum2 = sq1 + H1;
    float* sq2  = sum2 + H2;
    float* sc1  = sq2 + H2;
    float* sh1  = sc1 + H1;
    float* sc2  = sh1 + H1;
    float* sh2  = sc2 + H2;

    hipMemsetAsync(stats, 0, (size_t)(2 * H1 + 2 * H2) * sizeof(float), stream);

    knn_interp_concat<<<M_ / 256, 256, 0, stream>>>(xyz1, xyz2, points1, points2, Xc);

    gemm1_wmma<<<dim3(M_ / 128, H1 / 128), 256, 0, stream>>>(Xc, W1, b1, Y1);
    colstats<<<M_ / 256, H1, 0, stream>>>(Y1, H1, sum1, sq1);
    bnparams<<<1, 256, 0, stream>>>(sum1, sq1, g1, be1, sc1, sh1, H1);

    gemm2_wmma<<<dim3(M_ / 128, 1), 256, 0, stream>>>(Y1, W2, b2, sc1, sh1, Y2);
    colstats<<<M_ / 256, H2, 0, stream>>>(Y2, H2, sum2, sq2);
    bnparams<<<1, 256, 0, stream>>>(sum2, sq2, g2, be2, sc2, sh2, H2);

    bn_relu_transpose_out<<<dim3(M_ / 32, H2 / 32), 256, 0, stream>>>(Y2, sc2, sh2, out);
}